// NodePart_13365938225811
// MI455X (gfx1250) — compile-verified
//
#include <hip/hip_runtime.h>
#include <math.h>

typedef __attribute__((ext_vector_type(2))) float v2f;
typedef __attribute__((ext_vector_type(4))) float v4f;
typedef __attribute__((ext_vector_type(8))) float v8f;

#define N_COMS    8
#define MAX_NODES 10000
#define FEAT      128
#define ZDIM      256

// ---- ordered-uint encoding for float atomicMax (monotone map f32 -> u32) ----
__device__ __forceinline__ unsigned int enc_f32(float f) {
  unsigned int u = __float_as_uint(f);
  return (u & 0x80000000u) ? ~u : (u | 0x80000000u);
}
__device__ __forceinline__ float dec_f32(unsigned int k) {
  unsigned int u = (k & 0x80000000u) ? (k & 0x7fffffffu) : ~k;
  return __uint_as_float(u);
}

// Kernel 0: reset the 80 (chunk,community) max/denom slots every launch.
__global__ void k_init(unsigned int* __restrict__ colmax,
                       float* __restrict__ denom, int slots) {
  int i = blockIdx.x * blockDim.x + threadIdx.x;
  if (i < slots) {
    colmax[i] = enc_f32(-INFINITY);
    denom[i]  = 0.0f;
  }
}

// Kernel 1: phi = z @ W via V_WMMA_F32_16X16X4_F32, one wave per 16 nodes.
// W[d,c] = 1/32 iff d in [32c, 32c+32). Also reduces per-(chunk,c) column max.
__global__ void k_phi_wmma(const float* __restrict__ z,
                           float* __restrict__ phi,
                           unsigned int* __restrict__ colmax) {
  const int base = blockIdx.x * 16;          // first node of this wave's tile
  const int lane = threadIdx.x;              // 0..31 (wave32)
  const int half = lane >> 4;                // 0: K=k0,k0+1   1: K=k0+2,k0+3
  const int l    = lane & 15;                // M for A, N for B/C/D

  // A fragment source: row M=l of z, starting at K offset half*2
  const float* zr = z + (size_t)(base + l) * ZDIM + half * 2;

  v8f acc = {};
#pragma unroll
  for (int k0 = 0; k0 < ZDIM; k0 += 4) {
    v2f a;
    a.x = zr[k0];
    a.y = zr[k0 + 1];
    const int c = k0 >> 5;                   // community for this K-group
    const float bv = (l == c) ? 0.03125f : 0.0f;  // 1/32 averaging weight
    v2f b;
    b.x = bv;
    b.y = bv;                                // all 4 K-rows of B identical
    acc = __builtin_amdgcn_wmma_f32_16x16x4_f32(
        false, a, false, b, (short)0, acc, false, false);
  }

  // acc[r] = phi[node = base + 8*half + r][community = l] (l < 8 valid)
  if (l < N_COMS) {
    float lmax = -3.402823466e38f;
#pragma unroll
    for (int r = 0; r < 8; ++r) {
      const int node = base + half * 8 + r;
      phi[(size_t)node * N_COMS + l] = acc[r];
      lmax = fmaxf(lmax, acc[r]);
    }
    // combine the two node-halves (lanes l <-> l+16, both active here)
    lmax = fmaxf(lmax, __shfl_xor(lmax, 16, 32));
    if (half == 0) {
      const int chunk = base / MAX_NODES;    // tiles never straddle chunks
      atomicMax(&colmax[chunk * N_COMS + l], enc_f32(lmax));
    }
  }
}

// Kernel 2: e = exp(phi - colmax) (overwrites phi), reduce denom per (chunk,c).
__global__ void k_denom(float* __restrict__ phi,
                        const unsigned int* __restrict__ colmax,
                        float* __restrict__ denom) {
  const int chunk = blockIdx.y;
  const int local = blockIdx.x * blockDim.x + threadIdx.x;
  const int lane  = threadIdx.x & 31;
  const bool valid = local < MAX_NODES;

  float m[N_COMS];
#pragma unroll
  for (int c = 0; c < N_COMS; ++c) m[c] = dec_f32(colmax[chunk * N_COMS + c]);

  float e[N_COMS];
  const size_t nb = (size_t)(chunk * MAX_NODES + (valid ? local : 0)) * N_COMS;
#pragma unroll
  for (int c = 0; c < N_COMS; ++c) {
    e[c] = valid ? expf(phi[nb + c] - m[c]) : 0.0f;
    if (valid) phi[nb + c] = e[c];
  }
  // wave-level sum, one atomicAdd per wave per community
#pragma unroll
  for (int c = 0; c < N_COMS; ++c) {
    float v = e[c];
    for (int off = 16; off >= 1; off >>= 1) v += __shfl_xor(v, off, 32);
    if (lane == 0) atomicAdd(&denom[chunk * N_COMS + c], v);
  }
}

// Kernel 3: per node -> sm, s, nw, rowmax-mask; write mask + 8x expanded
// x_parts with non-temporal stores (412 MB streaming output).
__global__ void k_emit(const float* __restrict__ x,
                       const float* __restrict__ e_arr,
                       const float* __restrict__ denom,
                       float* __restrict__ out_x,
                       float* __restrict__ out_mask, int N) {
  const int wave = threadIdx.x >> 5;
  const int lane = threadIdx.x & 31;
  const int n = blockIdx.x * (blockDim.x >> 5) + wave;
  if (n >= N) return;                        // uniform per wave
  const int chunk = n / MAX_NODES;

  float sm = 0.0f;
  if (lane < N_COMS)
    sm = e_arr[(size_t)n * N_COMS + lane] / denom[chunk * N_COMS + lane];

  float s = sm;                              // lanes >= 8 contribute 0
  for (int off = 16; off >= 1; off >>= 1) s += __shfl_xor(s, off, 32);

  float nw = (lane < N_COMS) ? sm * (8.0f - s) : -3.402823466e38f;
  float rmax = nw;
  for (int off = 16; off >= 1; off >>= 1)
    rmax = fmaxf(rmax, __shfl_xor(rmax, off, 32));

  const bool sel = (lane < N_COMS) && (nw == rmax);
  if (lane < N_COMS)
    out_mask[(size_t)n * N_COMS + lane] = sel ? 1.0f : 0.0f;
  const unsigned int mbits = __builtin_amdgcn_ballot_w32(sel);

  // prefetch a future x row into cache (global_prefetch_b8)
  if (n + 64 < N) __builtin_prefetch(x + (size_t)(n + 64) * FEAT, 0, 1);

  const v4f xv = ((const v4f*)(x + (size_t)n * FEAT))[lane];
  const v4f zv = {0.0f, 0.0f, 0.0f, 0.0f};
#pragma unroll
  for (int c = 0; c < N_COMS; ++c) {
    v4f* dst = (v4f*)(out_x + ((size_t)c * N + n) * FEAT);
    __builtin_nontemporal_store(((mbits >> c) & 1u) ? xv : zv, &dst[lane]);
  }
}

extern "C" void kernel_launch(void* const* d_in, const int* in_sizes, int n_in,
                              void* d_out, int out_size, void* d_ws, size_t ws_size,
                              hipStream_t stream) {
  const float* x = (const float*)d_in[0];    // [N, 128] f32
  const float* z = (const float*)d_in[1];    // [N, 256] f32
  const int N = in_sizes[0] / FEAT;          // 100000
  const int n_chunks = N / MAX_NODES;        // 10
  const int slots = n_chunks * N_COMS;       // 80

  float* ws = (float*)d_ws;
  float* phi = ws;                                        // N*8 (phi, then e)
  unsigned int* colmax = (unsigned int*)(ws + (size_t)N * N_COMS);
  float* denom = (float*)(colmax + slots);

  float* out_x = (float*)d_out;                           // [8, N, 128]
  float* out_mask = out_x + (size_t)N_COMS * N * FEAT;    // [N, 8]

  k_init<<<1, 128, 0, stream>>>(colmax, denom, slots);

  k_phi_wmma<<<N / 16, 32, 0, stream>>>(z, phi, colmax);

  dim3 gC((MAX_NODES + 255) / 256, n_chunks);
  k_denom<<<gC, 256, 0, stream>>>(phi, colmax, denom);

  const int wpb = 8;  // waves per block
  k_emit<<<(N + wpb - 1) / wpb, 32 * wpb, 0, stream>>>(x, phi, denom, out_x,
                                                       out_mask, N);
}